// GCNModelAE_25769804171
// MI455X (gfx1250) — compile-verified
//
#include <hip/hip_runtime.h>

// ---------------------------------------------------------------------------
// GCN autoencoder forward on MI455X (gfx1250, wave32).
//
// Roofline: adj = z @ z.T writes 8192*8192*4 B = 256 MB (> 192 MB L2)
// -> HBM-store bound (~11 us @ 23.3 TB/s), stored with TH_STORE_NT.
// SpMM runs as CSR gather-sum (built on the fly) so there are ZERO f32
// atomics; all gather traffic (268 MB) hits the 192 MB L2.
// Dense math uses V_WMMA_F32_16X16X4_F32; adj waves are 2x4 register-blocked
// (32 wmma / wave) to cut z fragment re-reads from 512 MB to ~192 MB.
// ---------------------------------------------------------------------------

typedef float v2f __attribute__((ext_vector_type(2)));
typedef float v8f __attribute__((ext_vector_type(8)));

#define N_NODES 8192
#define N_EDGES 262144

// ---- f32 WMMA fragment helpers --------------------------------------------
// A 16x4 (f32): lanes 0-15 -> M=lane, hold K={0,1}; lanes 16-31 -> M=lane-16,
// hold K={2,3}. B 4x16 mirrored. C/D: VGPR i holds row M=i+8*half, col N=l.

template <bool RELU_A>
__device__ inline v8f wmma_tile_f32(const float* __restrict__ A,
                                    const float* __restrict__ B,
                                    int m0, int n0, int K, int lda, int ldb,
                                    int lane) {
  const int l    = lane & 15;
  const int half = lane >> 4;
  v8f c = {};
  const float* ap = A + (size_t)(m0 + l) * lda + 2 * half;
  const float* bp = B + (size_t)(2 * half) * ldb + n0 + l;
  for (int k0 = 0; k0 < K; k0 += 4) {
    v2f a, b;
    a.x = ap[k0];
    a.y = ap[k0 + 1];
    if (RELU_A) { a.x = fmaxf(a.x, 0.f); a.y = fmaxf(a.y, 0.f); }
    b.x = bp[(size_t)k0 * ldb];
    b.y = bp[(size_t)(k0 + 1) * ldb];
    c = __builtin_amdgcn_wmma_f32_16x16x4_f32(false, a, false, b,
                                              (short)0, c, false, false);
  }
  return c;
}

// ---- dense GEMM: C[M,N] = (relu?)(A[M,K]) @ B[K,N], one wave per tile -----
template <bool RELU_A>
__global__ void gemm_wmma(const float* __restrict__ A,
                          const float* __restrict__ B,
                          float* __restrict__ C,
                          int M, int K, int Ncols) {
  const int wid    = blockIdx.x * (blockDim.x >> 5) + (threadIdx.x >> 5);
  const int tilesN = Ncols >> 4;
  const int tm = wid / tilesN;
  const int tn = wid - tm * tilesN;
  if (tm * 16 >= M) return;                 // wave-uniform; EXEC stays all-1s
  const int lane = threadIdx.x & 31;
  const int l    = lane & 15;
  const int half = lane >> 4;
  v8f c = wmma_tile_f32<RELU_A>(A, B, tm * 16, tn * 16, K, K, Ncols, lane);
  float* p = C + (size_t)(tm * 16 + 8 * half) * Ncols + tn * 16 + l;
#pragma unroll
  for (int i = 0; i < 8; ++i) { *p = c[i]; p += Ncols; }
}

// ---- adj = z @ z.T : each wave computes a 2x4 group of 16x16 tiles --------
#define MT 2
#define NTILE 4
__global__ void adj_wmma(const float* __restrict__ Z, float* __restrict__ adj) {
  const int wid     = blockIdx.x * (blockDim.x >> 5) + (threadIdx.x >> 5);
  const int groupsN = (N_NODES / 16) / NTILE;      // 128
  const int gm = wid / groupsN;                    // 0..255
  const int gn = wid - gm * groupsN;
  const int m0 = gm * (16 * MT);
  const int n0 = gn * (16 * NTILE);
  const int lane = threadIdx.x & 31;
  const int l    = lane & 15;
  const int half = lane >> 4;

  v8f c[MT][NTILE] = {};
  const float* ap = Z + (size_t)(m0 + l) * 16 + 2 * half;   // A = z rows
  const float* bp = Z + (size_t)(n0 + l) * 16 + 2 * half;   // B = z^T cols
#pragma unroll
  for (int k0 = 0; k0 < 16; k0 += 4) {
    v2f a[MT], b[NTILE];
#pragma unroll
    for (int i = 0; i < MT; ++i) {
      a[i].x = ap[256 * i + k0];
      a[i].y = ap[256 * i + k0 + 1];
    }
#pragma unroll
    for (int j = 0; j < NTILE; ++j) {
      b[j].x = bp[256 * j + k0];
      b[j].y = bp[256 * j + k0 + 1];
    }
#pragma unroll
    for (int i = 0; i < MT; ++i)
#pragma unroll
      for (int j = 0; j < NTILE; ++j)
        c[i][j] = __builtin_amdgcn_wmma_f32_16x16x4_f32(
            false, a[i], false, b[j], (short)0, c[i][j], false, false);
  }
#pragma unroll
  for (int i = 0; i < MT; ++i)
#pragma unroll
    for (int j = 0; j < NTILE; ++j) {
      float* p = adj + (size_t)(m0 + 16 * i + 8 * half) * N_NODES
                     + n0 + 16 * j + l;
#pragma unroll
      for (int r = 0; r < 8; ++r) {
        __builtin_nontemporal_store(c[i][j][r], p);
        p += N_NODES;
      }
    }
}

// ---- CSR construction ------------------------------------------------------
__global__ void zero_i32(int* __restrict__ p, int n) {
  const int i = blockIdx.x * blockDim.x + threadIdx.x;
  if (i < n) p[i] = 0;
}

__global__ void degree_count(const int* __restrict__ row, int* __restrict__ deg) {
  const int e = blockIdx.x * blockDim.x + threadIdx.x;
  if (e < N_EDGES) atomicAdd(&deg[row[e]], 1);
}

// exclusive scan of deg[8192] -> row_ptr[8193]; single block of 1024 threads
__global__ void scan8192(const int* __restrict__ deg, int* __restrict__ row_ptr) {
  __shared__ int partial[1024];
  const int t = threadIdx.x;
  const int base = t * 8;
  int local[8];
  int s = 0;
#pragma unroll
  for (int i = 0; i < 8; ++i) { local[i] = s; s += deg[base + i]; }
  partial[t] = s;
  __syncthreads();
  for (int off = 1; off < 1024; off <<= 1) {
    const int v   = partial[t];
    const int add = (t >= off) ? partial[t - off] : 0;
    __syncthreads();
    partial[t] = v + add;
    __syncthreads();
  }
  const int pre = (t == 0) ? 0 : partial[t - 1];
#pragma unroll
  for (int i = 0; i < 8; ++i) row_ptr[base + i] = pre + local[i];
  if (t == 1023) row_ptr[8192] = partial[1023];
}

__global__ void copy_i32(const int* __restrict__ src, int* __restrict__ dst, int n) {
  const int i = blockIdx.x * blockDim.x + threadIdx.x;
  if (i < n) dst[i] = src[i];
}

__global__ void csr_fill(const int* __restrict__ row, int* __restrict__ cursor,
                         int* __restrict__ edge_id) {
  const int e = blockIdx.x * blockDim.x + threadIdx.x;
  if (e < N_EDGES) {
    const int slot = atomicAdd(&cursor[row[e]], 1);
    edge_id[slot] = e;
  }
}

// ---- CSR SpMM gather-sum: out[n] = sum_{e in row n} ew[e]*sup[col[e]] -----
// One thread per (node, float4 chunk); plain coalesced stores, no atomics.
template <int D>
__global__ void spmm_gather(const float* __restrict__ sup,
                            const int* __restrict__ row_ptr,
                            const int* __restrict__ edge_id,
                            const int* __restrict__ col,
                            const float* __restrict__ ew,
                            float* __restrict__ out) {
  const int chunks = D >> 2;
  const int gid = blockIdx.x * blockDim.x + threadIdx.x;
  const int n = gid / chunks;
  const int c = (gid - n * chunks) << 2;
  if (n >= N_NODES) return;
  const int beg = row_ptr[n];
  const int end = row_ptr[n + 1];
  float4 acc = {0.f, 0.f, 0.f, 0.f};
  for (int i = beg; i < end; ++i) {
    const int   e = edge_id[i];
    const float w = ew[e];
    const float4 v = *(const float4*)(sup + (size_t)col[e] * D + c);
    acc.x += v.x * w; acc.y += v.y * w; acc.z += v.z * w; acc.w += v.w * w;
  }
  *(float4*)(out + (size_t)n * D + c) = acc;
}

// ---------------------------------------------------------------------------
extern "C" void kernel_launch(void* const* d_in, const int* in_sizes, int n_in,
                              void* d_out, int out_size, void* d_ws, size_t ws_size,
                              hipStream_t stream) {
  const float* x   = (const float*)d_in[0];   // [8192,512]
  const float* w1  = (const float*)d_in[1];   // [512,256]
  const float* w2  = (const float*)d_in[2];   // [256,16]
  const float* ew  = (const float*)d_in[3];   // [E]
  const int*   row = (const int*)d_in[4];     // [E]
  const int*   col = (const int*)d_in[5];     // [E]
  float*       adj = (float*)d_out;           // [8192,8192]

  float* support1 = (float*)d_ws;                         // 8192*256
  float* h1       = support1 + (size_t)N_NODES * 256;     // 8192*256
  float* support2 = h1       + (size_t)N_NODES * 256;     // 8192*16
  float* z        = support2 + (size_t)N_NODES * 16;      // 8192*16
  int*   deg      = (int*)(z + (size_t)N_NODES * 16);     // 8192
  int*   row_ptr  = deg + N_NODES;                        // 8193
  int*   cursor   = row_ptr + N_NODES + 1;                // 8192
  int*   edge_id  = cursor + N_NODES;                     // E

  // --- CSR build (once per call; ~524K int atomics total) ---
  zero_i32<<<N_NODES / 256, 256, 0, stream>>>(deg, N_NODES);
  degree_count<<<N_EDGES / 256, 256, 0, stream>>>(row, deg);
  scan8192<<<1, 1024, 0, stream>>>(deg, row_ptr);
  copy_i32<<<N_NODES / 256, 256, 0, stream>>>(row_ptr, cursor, N_NODES);
  csr_fill<<<N_EDGES / 256, 256, 0, stream>>>(row, cursor, edge_id);

  // 1) support1 = x @ w1      (512*16 = 8192 wmma tiles)
  gemm_wmma<false><<<1024, 256, 0, stream>>>(x, w1, support1, N_NODES, 512, 256);

  // 2) h1 = segment_sum(support1[col] * ew)   (relu deferred to GEMM2 load)
  spmm_gather<256><<<(N_NODES * 64) / 256, 256, 0, stream>>>(
      support1, row_ptr, edge_id, col, ew, h1);

  // 3) support2 = relu(h1) @ w2   (512 wmma tiles)
  gemm_wmma<true><<<64, 256, 0, stream>>>(h1, w2, support2, N_NODES, 256, 16);

  // 4) z = segment_sum(support2[col] * ew)
  spmm_gather<16><<<(N_NODES * 4) / 256, 256, 0, stream>>>(
      support2, row_ptr, edge_id, col, ew, z);

  // 5) adj = z @ z.T : 2x4 tile groups, 32 wmma/wave, NT streaming stores
  adj_wmma<<<4096, 256, 0, stream>>>(z, adj);
}